// DGCNNAcc_24713241821962
// MI455X (gfx1250) — compile-verified
//
#include <hip/hip_runtime.h>
#include <hip/hip_bf16.h>
#include <math.h>

// ---------------------------------------------------------------------------
// DGCNN-style forward for MI455X (gfx1250, wave32, WMMA).
// All conv1x1 GEMMs run on V_WMMA_F32_16X16X4_F32 (fp32 A/B/C, matching the
// reference's fp32 numerics). Each wave computes a 64x16 output tile
// (4 stacked 16x16 WMMA tiles) so the B operand is reused 4x per K-step and
// the inner loop is branch-free (K padded to a multiple of 4, M multiple of 64).
// ---------------------------------------------------------------------------

typedef __attribute__((ext_vector_type(2))) float v2f;
typedef __attribute__((ext_vector_type(8))) float v8f;

#define BATCH 8
#define NPTS  2048
#define KPOOL 80   // K + 3P = 20 + 60

// ---------------------------------------------------------------------------
// prep: xc[b,c,n] = x[b,n,c] for c<3, row 3 zero-padded (so layer-1 K=4);
//       sq[b,n] = |x[b,n,:]|^2
// xc layout: (B, 4, N)
// ---------------------------------------------------------------------------
__global__ void prep_kernel(const float* __restrict__ x,
                            float* __restrict__ xc,
                            float* __restrict__ sq) {
    int i = blockIdx.x * blockDim.x + threadIdx.x;   // over B*N
    if (i >= BATCH * NPTS) return;
    int b = i / NPTS, n = i % NPTS;
    const float* p = x + (size_t)(b * NPTS + n) * 3;
    float a0 = p[0], a1 = p[1], a2 = p[2];
    float* xb = xc + (size_t)b * 4 * NPTS;
    xb[0 * NPTS + n] = a0;
    xb[1 * NPTS + n] = a1;
    xb[2 * NPTS + n] = a2;
    xb[3 * NPTS + n] = 0.0f;
    sq[i] = a0 * a0 + a1 * a1 + a2 * a2;
}

// ---------------------------------------------------------------------------
// pad_w: zero-pad (M, Kin) row-major weights to (M, Kout)
// ---------------------------------------------------------------------------
__global__ void pad_w_kernel(const float* __restrict__ w,
                             float* __restrict__ wp,
                             int M, int Kin, int Kout) {
    int i = blockIdx.x * blockDim.x + threadIdx.x;
    if (i >= M * Kout) return;
    int m = i / Kout, k = i % Kout;
    wp[i] = (k < Kin) ? w[m * Kin + k] : 0.0f;
}

// ---------------------------------------------------------------------------
// knn_topk: one workgroup (256 thr) per (b, n) row.
// pd[j] = 2*<x_n, x_j> - |x_n|^2 - |x_j|^2 held in LDS; 80 argmax rounds.
// Tie-break: lower index wins (matches lax.top_k ordering on ties).
// ---------------------------------------------------------------------------
__global__ void knn_topk_kernel(const float* __restrict__ x,
                                const float* __restrict__ sq,
                                int* __restrict__ idxpool) {
    __shared__ float dist[NPTS];
    __shared__ float rv[256];
    __shared__ int   ri[256];

    int bn = blockIdx.x;
    int b = bn / NPTS, n = bn % NPTS;
    const float* xb  = x  + (size_t)b * NPTS * 3;
    const float* sqb = sq + (size_t)b * NPTS;

    float qx = xb[n * 3 + 0], qy = xb[n * 3 + 1], qz = xb[n * 3 + 2];
    float qsq = sqb[n];

    for (int j = threadIdx.x; j < NPTS; j += blockDim.x) {
        float ix = xb[j * 3 + 0], iy = xb[j * 3 + 1], iz = xb[j * 3 + 2];
        float inner = qx * ix + qy * iy + qz * iz;
        dist[j] = 2.0f * inner - qsq - sqb[j];
    }
    __syncthreads();

    int* outp = idxpool + ((size_t)b * NPTS + n) * KPOOL;
    for (int t = 0; t < KPOOL; ++t) {
        float best = -INFINITY;
        int bidx = 0x7fffffff;
        for (int j = threadIdx.x; j < NPTS; j += blockDim.x) {
            float v = dist[j];
            if (v > best || (v == best && j < bidx)) { best = v; bidx = j; }
        }
        rv[threadIdx.x] = best;
        ri[threadIdx.x] = bidx;
        __syncthreads();
        for (int s = blockDim.x >> 1; s > 0; s >>= 1) {
            if (threadIdx.x < (unsigned)s) {
                float v2 = rv[threadIdx.x + s];
                int   i2 = ri[threadIdx.x + s];
                if (v2 > rv[threadIdx.x] ||
                    (v2 == rv[threadIdx.x] && i2 < ri[threadIdx.x])) {
                    rv[threadIdx.x] = v2; ri[threadIdx.x] = i2;
                }
            }
            __syncthreads();
        }
        if (threadIdx.x == 0) {
            outp[t] = ri[0];
            dist[ri[0]] = -INFINITY;
        }
        __syncthreads();
    }
}

// ---------------------------------------------------------------------------
// gemm_wmma: Y[b] = W (M x K) * X[b] (K x N), fp32, one 64x16 tile per wave.
// Requirements (guaranteed by caller): M % 64 == 0, K % 4 == 0, N % 16 == 0.
// V_WMMA_F32_16X16X4_F32 operand layouts (ISA 7.12.2):
//   A 16x4: lanes 0-15 hold K=k0,k0+1 for row M=lane; lanes 16-31 hold
//           K=k0+2,k0+3 (one v2f = one 64-bit load, rows of W are contiguous).
//   B 4x16: mirrored (row striped across lanes within a VGPR).
//   C/D   : 8 VGPRs; lanes 0-15 -> M=r, lanes 16-31 -> M=8+r.
// Early exit is wave-uniform => EXEC all-ones at every WMMA (ISA req).
// ---------------------------------------------------------------------------
__global__ void gemm_wmma_kernel(const float* __restrict__ W,
                                 const float* __restrict__ X,
                                 float* __restrict__ Y,
                                 int M, int K, int N, int B,
                                 long xBatchStride, long yBatchStride) {
    const int wavesPerBlock = blockDim.x >> 5;
    const int waveId = blockIdx.x * wavesPerBlock + (threadIdx.x >> 5);
    const int lane = threadIdx.x & 31;

    const int mTiles = M >> 6;           // 64-row macro tiles
    const int nTiles = N >> 4;
    const int tilesPerBatch = mTiles * nTiles;
    const int total = tilesPerBatch * B;
    if (waveId >= total) return;         // wave-uniform

    const int b  = waveId / tilesPerBatch;
    const int t  = waveId % tilesPerBatch;
    const int m0 = (t / nTiles) << 6;
    const int n0 = (t % nTiles) << 4;

    const float* Xb = X + (long)b * xBatchStride;
    float*       Yb = Y + (long)b * yBatchStride;

    const int half = lane >> 4;          // 0: lanes 0-15, 1: lanes 16-31
    const int l    = lane & 15;

    // B operand pointer: column n0+l, rows advance by 4 per K-step.
    const float* xp  = Xb + (long)(2 * half) * N + (n0 + l);
    // A operand pointers, one per 16-row subtile (contiguous K pairs).
    const float* wp0 = W + (long)(m0 +  0 + l) * K + 2 * half;
    const float* wp1 = W + (long)(m0 + 16 + l) * K + 2 * half;
    const float* wp2 = W + (long)(m0 + 32 + l) * K + 2 * half;
    const float* wp3 = W + (long)(m0 + 48 + l) * K + 2 * half;

    v8f acc0 = {}, acc1 = {}, acc2 = {}, acc3 = {};
    for (int k0 = 0; k0 < K; k0 += 4) {
        v2f bb;
        bb.x = xp[0];
        bb.y = xp[N];
        const v2f a0 = *(const v2f*)wp0;
        const v2f a1 = *(const v2f*)wp1;
        const v2f a2 = *(const v2f*)wp2;
        const v2f a3 = *(const v2f*)wp3;
        acc0 = __builtin_amdgcn_wmma_f32_16x16x4_f32(false, a0, false, bb, (short)0, acc0, false, false);
        acc1 = __builtin_amdgcn_wmma_f32_16x16x4_f32(false, a1, false, bb, (short)0, acc1, false, false);
        acc2 = __builtin_amdgcn_wmma_f32_16x16x4_f32(false, a2, false, bb, (short)0, acc2, false, false);
        acc3 = __builtin_amdgcn_wmma_f32_16x16x4_f32(false, a3, false, bb, (short)0, acc3, false, false);
        xp  += 4L * N;
        wp0 += 4; wp1 += 4; wp2 += 4; wp3 += 4;
    }

    const int n = n0 + l;
    float* yp = Yb + (long)(m0 + 8 * half) * N + n;
#pragma unroll
    for (int r = 0; r < 8; ++r) yp[(long)r * N] = acc0[r];
    yp += 16L * N;
#pragma unroll
    for (int r = 0; r < 8; ++r) yp[(long)r * N] = acc1[r];
    yp += 16L * N;
#pragma unroll
    for (int r = 0; r < 8; ++r) yp[(long)r * N] = acc2[r];
    yp += 16L * N;
#pragma unroll
    for (int r = 0; r < 8; ++r) yp[(long)r * N] = acc3[r];
}

// ---------------------------------------------------------------------------
// gn_stats: per (b, group) mean & population variance over (C/groups * N).
// grid = B*groups blocks of 256.
// ---------------------------------------------------------------------------
__global__ void gn_stats_kernel(const float* __restrict__ Y,
                                float* __restrict__ stats,
                                int C, int N, int groups) {
    __shared__ float s1[256], s2[256];
    const int bg = blockIdx.x;
    const int b = bg / groups, g = bg % groups;
    const int cpg = C / groups;
    const float* base = Y + (long)b * C * N + (long)g * cpg * N;
    const long total = (long)cpg * N;

    float s = 0.0f, ss = 0.0f;
    for (long i = threadIdx.x; i < total; i += blockDim.x) {
        float v = base[i];
        s += v; ss += v * v;
    }
    s1[threadIdx.x] = s; s2[threadIdx.x] = ss;
    __syncthreads();
    for (int r = blockDim.x >> 1; r > 0; r >>= 1) {
        if (threadIdx.x < (unsigned)r) {
            s1[threadIdx.x] += s1[threadIdx.x + r];
            s2[threadIdx.x] += s2[threadIdx.x + r];
        }
        __syncthreads();
    }
    if (threadIdx.x == 0) {
        float mean = s1[0] / (float)total;
        float var  = s2[0] / (float)total - mean * mean;
        stats[2 * bg + 0] = mean;
        stats[2 * bg + 1] = var;
    }
}

// ---------------------------------------------------------------------------
// gn_apply: y = (v - mean)*rsqrt(var+eps)*gamma[c] + beta[c]; LeakyReLU(0.2).
// transpose==0: Yout same (B,C,N) layout as Yin.
// transpose==1: Yout is (B,N,C) (final output).
// ---------------------------------------------------------------------------
__global__ void gn_apply_kernel(const float* __restrict__ Yin,
                                float* __restrict__ Yout,
                                const float* __restrict__ stats,
                                const float* __restrict__ gamma,
                                const float* __restrict__ beta,
                                int C, int N, int groups, int transpose) {
    const long i = (long)blockIdx.x * blockDim.x + threadIdx.x;
    const long totalElems = (long)BATCH * C * N;
    if (i >= totalElems) return;
    const int n = (int)(i % N);
    const int c = (int)((i / N) % C);
    const int b = (int)(i / ((long)C * N));
    const int cpg = C / groups;
    const int bg = b * groups + c / cpg;
    const float mean = stats[2 * bg + 0];
    const float var  = stats[2 * bg + 1];
    float v = Yin[i];
    v = (v - mean) * rsqrtf(var + 1e-5f) * gamma[c] + beta[c];
    v = (v > 0.0f) ? v : 0.2f * v;
    if (transpose) {
        Yout[((long)b * N + n) * C + c] = v;
    } else {
        Yout[i] = v;
    }
}

// ---------------------------------------------------------------------------
// gather_max: out[b, chOff+c, n] = max_k fa[b, c, idx[b,n,k]] + fb[b, c, n]
// One workgroup per (b,n); indices staged in LDS; channels across threads.
// fa/fb stay L2-resident (<= 2MB per batch) so the random gathers hit L2.
// ---------------------------------------------------------------------------
__global__ void gather_max_kernel(const float* __restrict__ fa,
                                  const float* __restrict__ fb,
                                  const int* __restrict__ idxpool,
                                  float* __restrict__ out,
                                  int C, int N, int kk,
                                  long outBatchStride, int outChOff) {
    __shared__ int sidx[KPOOL];
    const int bn = blockIdx.x;
    const int b = bn / N, n = bn % N;
    const int* ip = idxpool + ((long)b * N + n) * KPOOL;
    for (int t = threadIdx.x; t < kk; t += blockDim.x) sidx[t] = ip[t];
    __syncthreads();

    const float* fab = fa + (long)b * C * N;
    const float* fbb = fb + (long)b * C * N;
    float* ob = out + (long)b * outBatchStride + (long)outChOff * N;
    for (int c = threadIdx.x; c < C; c += blockDim.x) {
        const float* fc = fab + (long)c * N;
        float m = -INFINITY;
        for (int k = 0; k < kk; ++k) {
            float v = fc[sidx[k]];
            m = fmaxf(m, v);
        }
        ob[(long)c * N + n] = m + fbb[(long)c * N + n];
    }
}

// ---------------------------------------------------------------------------
// host-side helpers
// ---------------------------------------------------------------------------
static void launch_gemm(const float* W, const float* X, float* Y,
                        int M, int K, int N, long xbs, long ybs,
                        hipStream_t s) {
    const int mT = M >> 6, nT = N >> 4;
    const long total = (long)mT * nT * BATCH;
    const int wavesPerBlock = 8;  // 256 threads
    const long blocks = (total + wavesPerBlock - 1) / wavesPerBlock;
    gemm_wmma_kernel<<<dim3((unsigned)blocks), dim3(256), 0, s>>>(
        W, X, Y, M, K, N, BATCH, xbs, ybs);
}

static void launch_gn(const float* Yin, float* Yout, float* stats,
                      const float* gamma, const float* beta,
                      int C, int groups, int transpose, hipStream_t s) {
    gn_stats_kernel<<<dim3(BATCH * groups), dim3(256), 0, s>>>(Yin, stats, C, NPTS, groups);
    const long total = (long)BATCH * C * NPTS;
    const long blocks = (total + 255) / 256;
    gn_apply_kernel<<<dim3((unsigned)blocks), dim3(256), 0, s>>>(
        Yin, Yout, stats, gamma, beta, C, NPTS, groups, transpose);
}

extern "C" void kernel_launch(void* const* d_in, const int* in_sizes, int n_in,
                              void* d_out, int out_size, void* d_ws, size_t ws_size,
                              hipStream_t stream) {
    const float* x = (const float*)d_in[0];
    // weight triples: w,g,b starting at index 1
    const float* w1a = (const float*)d_in[1];  const float* g1a = (const float*)d_in[2];  const float* b1a = (const float*)d_in[3];
    const float* w1b = (const float*)d_in[4];  const float* g1b = (const float*)d_in[5];  const float* b1b = (const float*)d_in[6];
    const float* w2a = (const float*)d_in[7];  const float* g2a = (const float*)d_in[8];  const float* b2a = (const float*)d_in[9];
    const float* w2b = (const float*)d_in[10]; const float* g2b = (const float*)d_in[11]; const float* b2b = (const float*)d_in[12];
    const float* w3a = (const float*)d_in[13]; const float* g3a = (const float*)d_in[14]; const float* b3a = (const float*)d_in[15];
    const float* w3b = (const float*)d_in[16]; const float* g3b = (const float*)d_in[17]; const float* b3b = (const float*)d_in[18];
    const float* w4a = (const float*)d_in[19]; const float* g4a = (const float*)d_in[20]; const float* b4a = (const float*)d_in[21];
    const float* w4b = (const float*)d_in[22]; const float* g4b = (const float*)d_in[23]; const float* b4b = (const float*)d_in[24];
    const float* w51 = (const float*)d_in[25]; const float* g51 = (const float*)d_in[26]; const float* b51 = (const float*)d_in[27];
    const float* w52 = (const float*)d_in[28]; const float* g52 = (const float*)d_in[29]; const float* b52 = (const float*)d_in[30];

    // ---- workspace layout (bytes, 256B-aligned chunks) ----
    char* ws = (char*)d_ws;
    size_t off = 0;
    auto alloc = [&](size_t bytes) {
        void* p = ws + off;
        off += (bytes + 255) & ~(size_t)255;
        return p;
    };
    float* sq     = (float*)alloc(sizeof(float) * BATCH * NPTS);            //  64 KB
    float* xc     = (float*)alloc(sizeof(float) * BATCH * 4 * NPTS);        // 256 KB (K padded to 4)
    float* stats  = (float*)alloc(sizeof(float) * BATCH * 16 * 2);          //   1 KB
    float* w1aPad = (float*)alloc(sizeof(float) * 64 * 4);                  //   1 KB
    float* w1bPad = (float*)alloc(sizeof(float) * 64 * 4);                  //   1 KB
    int*   idxp   = (int*)  alloc(sizeof(int)   * (size_t)BATCH * NPTS * KPOOL); // 5.2 MB
    float* bufA   = (float*)alloc(sizeof(float) * (size_t)BATCH * 1024 * NPTS);  //  64 MB
    float* bufB   = (float*)alloc(sizeof(float) * (size_t)BATCH * 512  * NPTS);  //  32 MB
    float* xcat   = (float*)alloc(sizeof(float) * (size_t)BATCH * 512  * NPTS);  //  32 MB
    (void)ws_size; (void)n_in; (void)in_sizes; (void)out_size;

    const long XC_BS  = (long)4   * NPTS;
    const long CAT_BS = (long)512 * NPTS;

    // ---- prep: transpose+pad coords, squared norms, pad layer-1 weights ----
    prep_kernel<<<dim3((BATCH * NPTS + 255) / 256), dim3(256), 0, stream>>>(x, xc, sq);
    pad_w_kernel<<<dim3(1), dim3(256), 0, stream>>>(w1a, w1aPad, 64, 3, 4);
    pad_w_kernel<<<dim3(1), dim3(256), 0, stream>>>(w1b, w1bPad, 64, 3, 4);

    // ---- knn top-80 ----
    knn_topk_kernel<<<dim3(BATCH * NPTS), dim3(256), 0, stream>>>(x, sq, idxp);

    struct LayerDef {
        const float *wa, *ga, *ba, *wb, *gb, *bb;
        int cin, cout, groups, kk, chOff;
        const float* xin; long xinBS;
    } L[4] = {
        { w1aPad, g1a, b1a, w1bPad, g1b, b1b,   4,  64,  8, 20,   0, xc,              XC_BS  },
        { w2a,    g2a, b2a, w2b,    g2b, b2b,  64,  64,  8, 40,  64, xcat + 0*NPTS,   CAT_BS },
        { w3a,    g3a, b3a, w3b,    g3b, b3b,  64, 128,  8, 60, 128, xcat + 64*NPTS,  CAT_BS },
        { w4a,    g4a, b4a, w4b,    g4b, b4b, 128, 256, 16, 80, 256, xcat + 128*NPTS, CAT_BS },
    };

    for (int li = 0; li < 4; ++li) {
        const LayerDef& d = L[li];
        // branch a -> bufA (gathered), branch b -> bufB (pointwise add)
        launch_gemm(d.wa, d.xin, bufA, d.cout, d.cin, NPTS, d.xinBS, (long)d.cout * NPTS, stream);
        launch_gn(bufA, bufA, stats, d.ga, d.ba, d.cout, d.groups, 0, stream);
        launch_gemm(d.wb, d.xin, bufB, d.cout, d.cin, NPTS, d.xinBS, (long)d.cout * NPTS, stream);
        launch_gn(bufB, bufB, stats, d.gb, d.bb, d.cout, d.groups, 0, stream);
        gather_max_kernel<<<dim3(BATCH * NPTS), dim3(256), 0, stream>>>(
            bufA, bufB, idxp, xcat, d.cout, NPTS, d.kk, CAT_BS, d.chOff);
    }

    // ---- block 5_1: (B,512,N) -> (B,1024,N) ----
    launch_gemm(w51, xcat, bufA, 1024, 512, NPTS, CAT_BS, (long)1024 * NPTS, stream);
    launch_gn(bufA, bufA, stats, g51, b51, 1024, 16, 0, stream);

    // ---- block 5_2: (B,1024,N) -> (B,512,N), transposed to d_out (B,N,512) ----
    launch_gemm(w52, bufA, bufB, 512, 1024, NPTS, (long)1024 * NPTS, (long)512 * NPTS, stream);
    launch_gn(bufB, (float*)d_out, stats, g52, b52, 512, 16, 1, stream);
}